// CrossLayer_19902878449868
// MI455X (gfx1250) — compile-verified
//
#include <hip/hip_runtime.h>
#include <cstdint>
#include <cstddef>

// CrossLayer fused kernel for MI455X (gfx1250):
//   s = x1 @ w  (per-row dot, H=2048)
//   y = x0 * (1 + s)
//   out = LayerNorm(y) * gamma + beta   (eps = 1e-12)
//   d_out = concat(x0, out)
//
// Bandwidth-bound: 512 MB of HBM traffic -> ~22us floor @ 23.3 TB/s.
// CDNA5-specific paths used: global_load_async_to_lds_b128 (ASYNCcnt),
// s_wait_asynccnt, wave32 butterfly reductions, b128 non-temporal streams.

typedef float f32x4 __attribute__((ext_vector_type(4)));

static constexpr int kH     = 2048;
static constexpr int kBlock = 256;          // 8 wave32 per block
static constexpr int kPer   = kH / kBlock;  // 8 floats per thread
static constexpr float kLnEps = 1e-12f;

__device__ __forceinline__ float hsum4(f32x4 v) { return (v.x + v.y) + (v.z + v.w); }
__device__ __forceinline__ float sq4(f32x4 v) {
  return fmaf(v.x, v.x, fmaf(v.y, v.y, fmaf(v.z, v.z, v.w * v.w)));
}
__device__ __forceinline__ float dot4(f32x4 a, f32x4 b) {
  return fmaf(a.x, b.x, fmaf(a.y, b.y, fmaf(a.z, b.z, a.w * b.w)));
}
__device__ __forceinline__ f32x4 ln4(f32x4 x, f32x4 g, f32x4 b,
                                     float scale, float mean, float rstd) {
  f32x4 r;
  r.x = fmaf(fmaf(x.x, scale, -mean) * rstd, g.x, b.x);
  r.y = fmaf(fmaf(x.y, scale, -mean) * rstd, g.y, b.y);
  r.z = fmaf(fmaf(x.z, scale, -mean) * rstd, g.z, b.z);
  r.w = fmaf(fmaf(x.w, scale, -mean) * rstd, g.w, b.w);
  return r;
}

__global__ __launch_bounds__(kBlock) void cross_ln_fused(
    const float* __restrict__ x0, const float* __restrict__ x1,
    const float* __restrict__ weight, const float* __restrict__ gamma,
    const float* __restrict__ beta, float* __restrict__ out_x0,
    float* __restrict__ out_ln) {
  __shared__ float lds_x1[kH];
  __shared__ float red_dot[kBlock / 32];
  __shared__ float red_sum[kBlock / 32];
  __shared__ float red_sq[kBlock / 32];

  const int t = threadIdx.x;
  const int base = t * kPer;
  const size_t rowoff = (size_t)blockIdx.x * (size_t)kH;

  // ---- CDNA5 async copy: this thread's x1 chunk -> LDS (ASYNCcnt path) ----
  // the offset field is added to both the LDS and the global address
  const float* x1g = x1 + rowoff + base;
  unsigned lds_addr = (unsigned)(uintptr_t)(&lds_x1[base]);
  asm volatile("global_load_async_to_lds_b128 %0, %1, off"
               :: "v"(lds_addr), "v"(x1g) : "memory");
  asm volatile("global_load_async_to_lds_b128 %0, %1, off offset:16"
               :: "v"(lds_addr), "v"(x1g) : "memory");

  // Streaming read of x0 (held in registers, reused 3x: stats, copy-out, LN)
  const f32x4* x0v = (const f32x4*)(x0 + rowoff + base);
  f32x4 a0 = __builtin_nontemporal_load(x0v);
  f32x4 a1 = __builtin_nontemporal_load(x0v + 1);

  // L2-resident reused vectors
  const f32x4* wv = (const f32x4*)(weight + base);
  f32x4 w0 = wv[0], w1 = wv[1];

  // x0 first/second moments while the async engine fills LDS
  float psum = hsum4(a0) + hsum4(a1);
  float psq  = sq4(a0) + sq4(a1);

  // Wait for this wave's async LDS fills, then consume our own chunk
  asm volatile("s_wait_asynccnt 0" ::: "memory");
  f32x4 b0 = *(const f32x4*)(&lds_x1[base]);
  f32x4 b1 = *(const f32x4*)(&lds_x1[base + 4]);
  float pdot = dot4(b0, w0) + dot4(b1, w1);

  // wave32 butterfly reduction of (dot, sum, sumsq)
#pragma unroll
  for (int off = 16; off > 0; off >>= 1) {
    pdot += __shfl_xor(pdot, off, 32);
    psum += __shfl_xor(psum, off, 32);
    psq  += __shfl_xor(psq,  off, 32);
  }
  const int wave = t >> 5;
  if ((t & 31) == 0) {
    red_dot[wave] = pdot;
    red_sum[wave] = psum;
    red_sq[wave]  = psq;
  }
  __syncthreads();
  float dot = 0.f, sum = 0.f, sqs = 0.f;
#pragma unroll
  for (int i = 0; i < kBlock / 32; ++i) {
    dot += red_dot[i];
    sum += red_sum[i];
    sqs += red_sq[i];
  }

  // y = x0*(1+s): derive LN(y) stats from x0 moments
  const float scale = 1.0f + dot;
  const float inv_h = 1.0f / (float)kH;
  const float mean0 = sum * inv_h;
  const float var0  = fmaxf(fmaf(-mean0, mean0, sqs * inv_h), 0.0f);
  const float meanY = scale * mean0;
  const float rstd  = rsqrtf(fmaf(scale * scale, var0, kLnEps));

  const f32x4* gv = (const f32x4*)(gamma + base);
  const f32x4* bv = (const f32x4*)(beta + base);
  f32x4 g0 = gv[0], g1 = gv[1];
  f32x4 be0 = bv[0], be1 = bv[1];

  f32x4 o0 = ln4(a0, g0, be0, scale, meanY, rstd);
  f32x4 o1 = ln4(a1, g1, be1, scale, meanY, rstd);

  // Streaming writes: x0 passthrough + LN output
  f32x4* d0 = (f32x4*)(out_x0 + rowoff + base);
  f32x4* d1 = (f32x4*)(out_ln + rowoff + base);
  __builtin_nontemporal_store(a0, d0);
  __builtin_nontemporal_store(a1, d0 + 1);
  __builtin_nontemporal_store(o0, d1);
  __builtin_nontemporal_store(o1, d1 + 1);
}

extern "C" void kernel_launch(void* const* d_in, const int* in_sizes, int n_in,
                              void* d_out, int out_size, void* d_ws, size_t ws_size,
                              hipStream_t stream) {
  const float* x0 = (const float*)d_in[0];
  const float* x1 = (const float*)d_in[1];
  const float* w  = (const float*)d_in[2];
  const float* g  = (const float*)d_in[3];
  const float* b  = (const float*)d_in[4];

  const int rows = in_sizes[0] / kH;  // B = 16384
  float* out_x0 = (float*)d_out;                        // first B*H: x0 copy
  float* out_ln = out_x0 + (size_t)rows * (size_t)kH;   // second B*H: LN out

  cross_ln_fused<<<dim3(rows), dim3(kBlock), 0, stream>>>(x0, x1, w, g, b,
                                                          out_x0, out_ln);
  (void)n_in; (void)out_size; (void)d_ws; (void)ws_size;
}